// GaussianMRIModel_23862838296739
// MI455X (gfx1250) — compile-verified
//
#include <hip/hip_runtime.h>
#include <hip/hip_bf16.h>
#include <stdint.h>

typedef __attribute__((ext_vector_type(2))) float v2f;
typedef __attribute__((ext_vector_type(8))) float v8f;

#define GD 160
#define GH 160
#define GW 160
#define PATCH 20
#define REC_WORDS 24          // 96 bytes per point, 16B-aligned
#define KLOG2E 0.7213475204444817f   // 0.5 * log2(e)

// ---------------------------------------------------------------------------
// Kernel A: per-point preprocessing (quat->R, precision matrix, bbox, valid)
// ---------------------------------------------------------------------------
__global__ void gm_prep(const float* __restrict__ centers,
                        const float* __restrict__ log_scales,
                        const float* __restrict__ quats,
                        const float* __restrict__ rho,
                        float* __restrict__ rec, int n) {
  int i = blockIdx.x * blockDim.x + threadIdx.x;
  if (i >= n) return;

  float qw = quats[i*4+0], qx = quats[i*4+1], qy = quats[i*4+2], qz = quats[i*4+3];
  float nrm = fmaxf(sqrtf(qw*qw + qx*qx + qy*qy + qz*qz), 1e-6f);
  float inr = 1.0f / nrm;
  qw *= inr; qx *= inr; qy *= inr; qz *= inr;
  float ww = qw*qw, xx = qx*qx, yy = qy*qy, zz = qz*qz;
  float wx = qw*qx, wy = qw*qy, wz = qw*qz;
  float xy = qx*qy, xz = qx*qz, yz = qy*qz;
  float R00 = ww+xx-yy-zz, R01 = 2.f*(xy-wz), R02 = 2.f*(xz+wy);
  float R10 = 2.f*(xy+wz), R11 = ww-xx+yy-zz, R12 = 2.f*(yz-wx);
  float R20 = 2.f*(xz-wy), R21 = 2.f*(yz+wx), R22 = ww-xx-yy+zz;

  float s0 = __expf(log_scales[i*3+0]);
  float s1 = __expf(log_scales[i*3+1]);
  float s2 = __expf(log_scales[i*3+2]);
  float c0 = fmaxf(s0, 1e-4f), c1 = fmaxf(s1, 1e-4f), c2 = fmaxf(s2, 1e-4f);
  float i0 = 1.0f/(c0*c0), i1 = 1.0f/(c1*c1), i2 = 1.0f/(c2*c2);

  // prec = R * diag(inv) * R^T
  float p00 = R00*R00*i0 + R01*R01*i1 + R02*R02*i2;
  float p11 = R10*R10*i0 + R11*R11*i1 + R12*R12*i2;
  float p22 = R20*R20*i0 + R21*R21*i1 + R22*R22*i2;
  float p01 = R00*R10*i0 + R01*R11*i1 + R02*R12*i2;
  float p02 = R00*R20*i0 + R01*R21*i1 + R02*R22*i2;
  float p12 = R10*R20*i0 + R11*R21*i1 + R12*R22*i2;

  float cvz = centers[i*3+0] * (float)(GD-1);
  float cvy = centers[i*3+1] * (float)(GH-1);
  float cvx = centers[i*3+2] * (float)(GW-1);

  float rad = fmaxf(fmaxf(s0, s1), s2) * 3.0f;
  int bz = (int)floorf(cvz - rad), by = (int)floorf(cvy - rad), bx = (int)floorf(cvx - rad);
  int hz = (int)ceilf(cvz + rad),  hy = (int)ceilf(cvy + rad),  hx = (int)ceilf(cvx + rad);

  float r0 = rho[i*2+0], r1 = rho[i*2+1];
  float amp = sqrtf(r0*r0 + r1*r1);
  int valid = (amp >= 1e-6f) && (rad >= 1e-3f);

  float* o = rec + (size_t)i * REC_WORDS;
  o[0] = cvz; o[1] = cvy; o[2] = cvx;
  o[3] = p00; o[4] = p11; o[5] = p22;
  o[6] = p01; o[7] = p02; o[8] = p12;
  o[9] = r0;  o[10] = r1; o[11] = 0.0f;
  o[12] = __int_as_float(bz); o[13] = __int_as_float(by); o[14] = __int_as_float(bx);
  o[15] = __int_as_float(valid);
  o[16] = __int_as_float(hz); o[17] = __int_as_float(hy); o[18] = __int_as_float(hx);
  o[19] = 0.f; o[20] = 0.f; o[21] = 0.f; o[22] = 0.f; o[23] = 0.f;
}

// ---------------------------------------------------------------------------
// Kernel B: zero the output volume (poisoned by the harness)
// ---------------------------------------------------------------------------
__global__ void gm_zero(float4* __restrict__ out, int n4) {
  int i = blockIdx.x * blockDim.x + threadIdx.x;
  if (i < n4) out[i] = make_float4(0.f, 0.f, 0.f, 0.f);
}

// ---------------------------------------------------------------------------
// Kernel C: one block (12 wave32s) per Gaussian.
//  1. async-copy the 96B param record global->LDS (ASYNCcnt path)
//  2. 12 WMMAs (V_WMMA_F32_16X16X4_F32) build the three 20x20 pairwise
//     tables; the -0.5*log2(e) exp-scale is folded into the WMMA operands
//     so D feeds a bare v_exp_f32 (hardware exp2) directly
//  3. 8000-voxel splat: 2 LDS-product mults + 2 global f32 atomics per voxel
// ---------------------------------------------------------------------------
__global__ void __launch_bounds__(384) gm_splat(const float* __restrict__ rec,
                                                float* __restrict__ out) {
  __shared__ float tbl[3][32 * 32];
  __shared__ __align__(16) float prm[REC_WORDS];

  const int tid = threadIdx.x;
  const float* r = rec + (size_t)blockIdx.x * REC_WORDS;

  // --- (1) async load of the point record into LDS -------------------------
  if (tid < 6) {
    unsigned ldso = (unsigned)(uintptr_t)(&prm[0]) + (unsigned)tid * 16u;
    unsigned long long ga = (unsigned long long)(uintptr_t)(r + tid * 4);
    asm volatile("global_load_async_to_lds_b128 %0, %1, off"
                 :: "v"(ldso), "v"(ga) : "memory");
  }
  if (tid < 32) {
    asm volatile("s_wait_asynccnt 0" ::: "memory");
  }
  __syncthreads();

  const int valid = __float_as_int(prm[15]);
  if (!valid) return;  // uniform per block

  const float cvz = prm[0], cvy = prm[1], cvx = prm[2];
  const float p00 = prm[3], p11 = prm[4], p22 = prm[5];
  const float p01 = prm[6], p02 = prm[7], p12 = prm[8];
  const float rho0 = prm[9], rho1 = prm[10];
  const int bz = __float_as_int(prm[12]);
  const int by = __float_as_int(prm[13]);
  const int bx = __float_as_int(prm[14]);
  const int hz = __float_as_int(prm[16]);
  const int hy = __float_as_int(prm[17]);
  const int hx = __float_as_int(prm[18]);

  // --- (2) pairwise exponent tables via WMMA -------------------------------
  // wave w of 12: mat = w>>2 in {ZY, ZX, YX}; tile (ti,tj) of the 2x2 tiling
  // WMMA computes F[m,n] = -KLOG2E*(pr*dr^2/2 + pc*dc^2/2 + 2*px*dr*dc)
  // as a K=4 matmul:  A row m = [ -K/2*pr*dr^2 , dr , 1 , 0 ]
  //                   B col n = [ 1 , -2K*px*dc , -K/2*pc*dc^2 , 0 ]
  // so table entry = exp2(F) = exp(-0.5*e).
  {
    const int wave = tid >> 5, lane = tid & 31;
    const int mat = wave >> 2, t = wave & 3, ti = t >> 1, tj = t & 1;

    float cr, cc, pr, pc, px; int br, bc;
    if (mat == 0)      { cr = cvz; cc = cvy; pr = p00; pc = p11; px = p01; br = bz; bc = by; }
    else if (mat == 1) { cr = cvz; cc = cvx; pr = p00; pc = p22; px = p02; br = bz; bc = bx; }
    else               { cr = cvy; cc = cvx; pr = p11; pc = p22; px = p12; br = by; bc = bx; }

    const float prk = -0.5f * KLOG2E * pr;   // scaled diagonal (rows)
    const float pck = -0.5f * KLOG2E * pc;   // scaled diagonal (cols)
    const float pxk = -2.0f * KLOG2E * px;   // scaled cross term

    const int mlo = lane & 15;
    // A (16x4): lanes 0-15 hold K=0,1 (VGPR0,1); lanes 16-31 hold K=2,3
    float a0, a1;
    if (lane < 16) {
      float dr = (float)(br + ti * 16 + mlo) - cr;
      a0 = prk * dr * dr;
      a1 = dr;
    } else {
      a0 = 1.0f;   // K=2 column of ones
      a1 = 0.0f;   // K=3 zero pad
    }
    // B (4x16): VGPR0: lanes 0-15 = K0 row, lanes 16-31 = K2 row; VGPR1: K1/K3
    float dc = (float)(bc + tj * 16 + mlo) - cc;
    float b0v = (lane < 16) ? 1.0f : pck * dc * dc;
    float b1v = (lane < 16) ? pxk * dc : 0.0f;

    v2f A = { a0, a1 };
    v2f B = { b0v, b1v };
    v8f C = {};
    v8f Dm = __builtin_amdgcn_wmma_f32_16x16x4_f32(
        /*neg_a=*/false, A, /*neg_b=*/false, B,
        /*c_mod=*/(short)0, C, /*reuse_a=*/false, /*reuse_b=*/false);

    const int nofs = tj * 16 + mlo;
    const int mbase = ti * 16 + ((lane < 16) ? 0 : 8);
#pragma unroll
    for (int rr = 0; rr < 8; ++rr) {
      // raw v_exp_f32: flush-to-zero tail is exactly right for a Gaussian;
      // upper clamp keeps the unread padded rows finite (exp2(115) ~ 4e34)
      tbl[mat][(mbase + rr) * 32 + nofs] =
          __builtin_amdgcn_exp2f(fminf(Dm[rr], 115.0f));
    }
  }
  __syncthreads();

  // --- (3) splat the 20^3 patch -------------------------------------------
  // Incremental base-20 digit decode: stride 384 = (z,y,x) digits (0,19,4).
  {
    int x = tid % PATCH;
    int rem = tid / PATCH;
    int y = rem % PATCH;
    int z = rem / PATCH;
    for (int t = tid; t < PATCH * PATCH * PATCH; t += 384) {
      const int iz = bz + z, iy = by + y, ix = bx + x;
      const bool inb = (iz >= 0) & (iz < GD) & (iz <= hz) &
                       (iy >= 0) & (iy < GH) & (iy <= hy) &
                       (ix >= 0) & (ix < GW) & (ix <= hx);
      if (inb) {
        const float w = tbl[0][z * 32 + y] * tbl[1][z * 32 + x] * tbl[2][y * 32 + x];
        const int lin = (iz * GH + iy) * GW + ix;
        atomicAdd(&out[2 * lin + 0], w * rho0);
        atomicAdd(&out[2 * lin + 1], w * rho1);
      }
      // advance digits by (0,19,4) with carries (y <= 39 before carry)
      x += 4; y += 19;
      if (x >= PATCH) { x -= PATCH; y += 1; }
      if (y >= PATCH) { y -= PATCH; z += 1; }
    }
  }
}

// ---------------------------------------------------------------------------
extern "C" void kernel_launch(void* const* d_in, const int* in_sizes, int n_in,
                              void* d_out, int out_size, void* d_ws, size_t ws_size,
                              hipStream_t stream) {
  const float* centers    = (const float*)d_in[0];
  const float* log_scales = (const float*)d_in[1];
  const float* quats      = (const float*)d_in[2];
  const float* rho        = (const float*)d_in[3];
  float* out = (float*)d_out;
  float* rec = (float*)d_ws;              // n * 24 floats (96 B/point)

  const int n = in_sizes[0] / 3;          // 25000

  // A: prep per-point records
  gm_prep<<<(n + 255) / 256, 256, 0, stream>>>(centers, log_scales, quats, rho, rec, n);

  // B: zero output volume (160^3 * 2 floats, divisible by 4)
  const int n4 = out_size / 4;
  gm_zero<<<(n4 + 255) / 256, 256, 0, stream>>>((float4*)out, n4);

  // C: one block of 12 wave32s per Gaussian
  gm_splat<<<n, 384, 0, stream>>>(rec, out);
}